// RelKDAdapter_89378269430325
// MI455X (gfx1250) — compile-verified
//
#include <hip/hip_runtime.h>

typedef __attribute__((ext_vector_type(2))) float v2f;
typedef __attribute__((ext_vector_type(8))) float v8f;

#define D_IN  256   // D_SRC
#define D_REL 128

// ---------------------------------------------------------------- zero fill
__global__ void zero_f32(float* __restrict__ p, long n) {
  long i = (long)blockIdx.x * blockDim.x + threadIdx.x;
  long stride = (long)gridDim.x * blockDim.x;
  for (; i < n; i += stride) p[i] = 0.0f;
}

// ---------------------------------------------------------------- GEMM via fp32 WMMA
// C[M x 128] = A[M x 256] * B[256 x 128], all row-major fp32.
// One wave per 16x16 output tile; K consumed 4 at a time by v_wmma_f32_16x16x4_f32.
__global__ __launch_bounds__(256) void gemm_wmma_f32(
    const float* __restrict__ A, const float* __restrict__ B,
    float* __restrict__ C, int M) {
  const int tilesN = D_REL / 16;                      // 8
  int wave = blockIdx.x * (blockDim.x >> 5) + (threadIdx.x >> 5);
  int lane = threadIdx.x & 31;
  int tm = wave / tilesN;
  int tn = wave - tm * tilesN;
  if (tm * 16 >= M) return;                           // wave-uniform: EXEC stays all-1s

  int half = lane >> 4;                               // 0: K lo pair, 1: K hi pair
  int l15  = lane & 15;

  // A fragment: lane holds A[m][k+2*half], A[m][k+2*half+1]
  const float* Arow = A + (size_t)(tm * 16 + l15) * D_IN + half * 2;
  // B fragment: lane holds B[k+2*half][n], B[k+2*half+1][n]
  const float* Bcol = B + tn * 16 + l15;

  v8f acc = {};
#pragma unroll 8
  for (int k = 0; k < D_IN; k += 4) {
    v2f a = *(const v2f*)(Arow + k);                  // contiguous pair, 8B aligned
    int kb = k + half * 2;
    v2f b;
    b.x = Bcol[(size_t)(kb + 0) * D_REL];
    b.y = Bcol[(size_t)(kb + 1) * D_REL];
    // 8 args: (neg_a, A, neg_b, B, c_mod, C, reuse_a, reuse_b)
    acc = __builtin_amdgcn_wmma_f32_16x16x4_f32(
        false, a, false, b, (short)0, acc, false, false);
  }

  // C/D layout: VGPR r -> row (tm*16 + 8*half + r), col (tn*16 + l15)
  float* Cp = C + (size_t)(tm * 16 + half * 8) * D_REL + tn * 16 + l15;
#pragma unroll
  for (int r = 0; r < 8; ++r) Cp[(size_t)r * D_REL] = acc[r];
}

// ---------------------------------------------------------------- degree count
__global__ void deg_count(const int* __restrict__ edst, float* __restrict__ deg, int E) {
  int e = blockIdx.x * blockDim.x + threadIdx.x;
  if (e < E) atomicAdd(&deg[edst[e]], 1.0f);
}

// ---------------------------------------------------------------- edge scatter-add
// One wave per edge: lane l moves cols [4l, 4l+4) of p_src[src] into agg[dst].
// p_src (51MB) + agg (51MB) are L2-resident (192MB L2), so atomics run at L2 rate.
__global__ __launch_bounds__(256) void scatter_add(
    const float* __restrict__ P, const int* __restrict__ esrc,
    const int* __restrict__ edst, float* __restrict__ agg, int E) {
  long t = (long)blockIdx.x * blockDim.x + threadIdx.x;
  int e = (int)(t >> 5);
  if (e >= E) return;
  int lane = (int)(t & 31);
  int s = esrc[e];
  int d = edst[e];
  const float4 v = *(const float4*)(P + (size_t)s * D_REL + lane * 4);
  float* dp = agg + (size_t)d * D_REL + lane * 4;
  atomicAdd(dp + 0, v.x);
  atomicAdd(dp + 1, v.y);
  atomicAdd(dp + 2, v.z);
  atomicAdd(dp + 3, v.w);
}

// ---------------------------------------------------------------- normalize + clamp deg
__global__ void finalize_div(float* __restrict__ dst, float* __restrict__ deg, int N) {
  long t = (long)blockIdx.x * blockDim.x + threadIdx.x;
  int i = (int)(t >> 7);                              // row  (D_REL = 128)
  if (i >= N) return;
  int c = (int)(t & 127);
  float dg = deg[i];
  float dc = fmaxf(dg, 1.0f);                         // idempotent: safe vs c==0 write
  dst[t] = dst[t] / dc;
  if (c == 0) deg[i] = dc;                            // output deg is the clamped one
}

// ---------------------------------------------------------------- launch
extern "C" void kernel_launch(void* const* d_in, const int* in_sizes, int n_in,
                              void* d_out, int out_size, void* d_ws, size_t ws_size,
                              hipStream_t stream) {
  const float* x_src    = (const float*)d_in[0];
  // d_in[1] = x_dst, d_in[3] = W_dst: unused (p_dst is discarded by reference)
  const float* W_src    = (const float*)d_in[2];
  const int*   edge_src = (const int*)d_in[4];
  const int*   edge_dst = (const int*)d_in[5];

  const int N_src = in_sizes[0] / D_IN;
  const int N_dst = in_sizes[1] / D_IN;
  const int E     = in_sizes[4];

  float* out   = (float*)d_out;
  float* dst   = out;                                          // [N_dst, 128]
  float* p_src = out + (size_t)N_dst * D_REL;                  // [N_src, 128]
  float* deg   = p_src + (size_t)N_src * D_REL;                // [N_dst]

  // 1) zero the atomic accumulators (dst region + deg region)
  zero_f32<<<2048, 256, 0, stream>>>(dst, (long)N_dst * D_REL);
  zero_f32<<<256, 256, 0, stream>>>(deg, (long)N_dst);

  // 2) p_src = x_src @ W_src   (fp32 WMMA)
  int tilesM = (N_src + 15) / 16;
  int waves  = tilesM * (D_REL / 16);
  gemm_wmma_f32<<<(waves + 7) / 8, 256, 0, stream>>>(x_src, W_src, p_src, N_src);

  // 3) deg = segment_sum(ones, edge_dst)
  deg_count<<<(E + 255) / 256, 256, 0, stream>>>(edge_dst, deg, E);

  // 4) dst += p_src[edge_src] scattered to edge_dst (one wave per edge)
  long sthreads = (long)E * 32;
  scatter_add<<<(int)((sthreads + 255) / 256), 256, 0, stream>>>(
      p_src, edge_src, edge_dst, dst, E);

  // 5) dst /= max(deg,1); deg = max(deg,1)
  long fthreads = (long)N_dst * D_REL;
  finalize_div<<<(int)((fthreads + 255) / 256), 256, 0, stream>>>(dst, deg, N_dst);
}